// GVPPairEnergies_76527727280582
// MI455X (gfx1250) — compile-verified
//
// GVP-GNN pair-energy forward for MI455X (gfx1250), fp32 throughout.
//
// Design notes (MI455X reasoning):
//  * Total math ~20 GFLOP fp32, total HBM traffic ~150 MB; working set fits in
//    the 192 MB L2, so the kernel sequence is latency-bound, not BW-bound.
//    We keep reference fp32 precision and use CDNA5's native fp32 matrix op
//    V_WMMA_F32_16X16X4_F32 for every GEMM-shaped computation.
//  * GEMM kernel: one 64x16 C strip per wave32 (4 accumulators share one B
//    fragment -> 4x B reuse, 4 independent wmmas back-to-back for ILP).
//    K-body loop is guard-free (clamped addressing; junk rows/cols only feed
//    store-masked lanes); only the ragged-K tail uses selects, once per wave.
//  * Vector features stored channel-major [row][3][nv] so the geometric
//    einsums are plain dense GEMMs with contiguous rows.
//  * Irregular work (top-K select, dihedrals, RBF/PE, layernorms, gathers,
//    sigmoid vector gates, pair merge) runs in small scalar wave32 kernels.

#include <hip/hip_runtime.h>

#define DEV_INLINE __device__ __forceinline__

namespace {

constexpr int Nn      = 1536;
constexpr int KN      = 30;
constexpr int Esz     = Nn * KN;       // 46080 edges
constexpr int IN_DIM  = 64;
constexpr int HS      = 50;
constexpr int HV      = 8;
constexpr int OUT_DIM = 400;
constexpr float EPSf  = 1e-8f;

typedef float v2f __attribute__((ext_vector_type(2)));
typedef float v8f __attribute__((ext_vector_type(8)));

// ---------------------------------------------------------------------------
// fp32 GEMM via V_WMMA_F32_16X16X4_F32.
// C[M,Ncols] = act(A[M,K] @ W[K,Ncols] + bias)   (all row-major)
// One 64x16 C strip per wave: 4 sub-tiles (16x16) share each B fragment.
// A frag (16x4): lanes 0-15 rows hold K={kk,kk+1}; lanes 16-31 K={kk+2,kk+3}.
// B frag (4x16): VGPR0 = row kk (lanes0-15)/kk+2 (lanes16-31), VGPR1 = kk+1/kk+3.
// C tile: vgpr r = row r (lanes0-15) / r+8 (lanes16-31), col = lane&15.
// Out-of-range rows/cols use clamped (in-bounds) addresses; their junk lands
// only in store-masked lanes.  Ragged K handled by one select-guarded tail.
// ---------------------------------------------------------------------------
__global__ __launch_bounds__(256) void gemm_f32_wmma(
    const float* __restrict__ A, int lda,
    const float* __restrict__ W, int ldw,
    float* __restrict__ C, int ldc,
    const float* __restrict__ bias,
    int M, int K, int Ncols, int act)
{
  const int lane = threadIdx.x & 31;
  const int wave = threadIdx.x >> 5;
  const int tilesN  = (Ncols + 15) >> 4;
  const int tilesM  = (M + 15) >> 4;
  const int stripsM = (tilesM + 3) >> 2;           // 4 M-tiles per wave
  const int nWork   = stripsM * tilesN;
  const int work = blockIdx.x * 8 + wave;          // wave-uniform
  if (work >= nWork) return;                       // uniform exit, EXEC all-ones
  const int sm = work / tilesN, tn = work % tilesN;
  const int half = lane >> 4, l16 = lane & 15;
  const int khalf = 2 * half;

  const int col  = tn * 16 + l16;
  const int colc = (col < Ncols) ? col : (Ncols - 1);   // clamped B/bias column

  const float* aRow[4];
#pragma unroll
  for (int r = 0; r < 4; ++r) {
    const int row = (sm * 4 + r) * 16 + l16;
    const int rowc = (row < M) ? row : (M - 1);         // clamped A row
    aRow[r] = A + (size_t)rowc * lda;
  }
  const float* wCol = W + colc;

  v8f acc[4];
#pragma unroll
  for (int r = 0; r < 4; ++r) acc[r] = v8f{0.f,0.f,0.f,0.f,0.f,0.f,0.f,0.f};

  const int Kbody = K & ~3;
  for (int kk = 0; kk < Kbody; kk += 4) {
    v2f b;
    b.x = wCol[(size_t)(kk + khalf) * ldw];
    b.y = wCol[(size_t)(kk + khalf + 1) * ldw];
#pragma unroll
    for (int r = 0; r < 4; ++r) {
      v2f a;
      a.x = aRow[r][kk + khalf];
      a.y = aRow[r][kk + khalf + 1];
      acc[r] = __builtin_amdgcn_wmma_f32_16x16x4_f32(false, a, false, b,
                                                     (short)0, acc[r], false, false);
    }
  }
  if (Kbody < K) {                                  // ragged-K tail (once)
    const int k0 = Kbody + khalf;
    const bool ok0 = (k0 < K), ok1 = (k0 + 1 < K);
    const int k0c = ok0 ? k0 : (K - 1);
    const int k1c = ok1 ? (k0 + 1) : (K - 1);
    v2f b;                                          // B needs no select (see note)
    b.x = wCol[(size_t)k0c * ldw];
    b.y = wCol[(size_t)k1c * ldw];
#pragma unroll
    for (int r = 0; r < 4; ++r) {
      const float a0 = aRow[r][k0c];
      const float a1 = aRow[r][k1c];
      v2f a;
      a.x = ok0 ? a0 : 0.f;                         // zero A kills OOB-k term
      a.y = ok1 ? a1 : 0.f;
      acc[r] = __builtin_amdgcn_wmma_f32_16x16x4_f32(false, a, false, b,
                                                     (short)0, acc[r], false, false);
    }
  }

#pragma unroll
  for (int r = 0; r < 4; ++r) {
    const int tmr = sm * 4 + r;
    if (tmr >= tilesM) continue;
#pragma unroll
    for (int rr = 0; rr < 8; ++rr) {
      const int orow = tmr * 16 + rr + 8 * half;
      if (orow < M && col < Ncols) {
        float v = acc[r][rr];
        if (bias) v += bias[col];
        if (act == 1) v = fmaxf(v, 0.f);
        C[(size_t)orow * ldc + col] = v;
      }
    }
  }
}

// ---------------------------------------------------------------------------
// small geometry helpers
// ---------------------------------------------------------------------------
DEV_INLINE void norm3(float* v) {
  float n = sqrtf(fmaxf(v[0]*v[0] + v[1]*v[1] + v[2]*v[2], EPSf));
  v[0] /= n; v[1] /= n; v[2] /= n;
}
DEV_INLINE void cross3(const float* a, const float* b, float* o) {
  o[0] = a[1]*b[2] - a[2]*b[1];
  o[1] = a[2]*b[0] - a[0]*b[2];
  o[2] = a[0]*b[1] - a[1]*b[0];
}
DEV_INLINE float bbAtom(const float* X, int p, int c) {  // flattened (N,CA,C)
  const int res = p / 3, a = p % 3;
  return X[(size_t)res * 12 + a * 3 + c];
}
DEV_INLINE void udir(const float* X, int p, float* u) {
  u[0] = bbAtom(X, p + 1, 0) - bbAtom(X, p, 0);
  u[1] = bbAtom(X, p + 1, 1) - bbAtom(X, p, 1);
  u[2] = bbAtom(X, p + 1, 2) - bbAtom(X, p, 2);
  norm3(u);
}

// ---------------------------------------------------------------------------
// top-K nearest neighbors per node (matches jax.lax.top_k of -(D+pen):
// ascending distance, ties broken by smaller index).  One block per node.
// ---------------------------------------------------------------------------
__global__ __launch_bounds__(128) void topk_kernel(
    const float* __restrict__ X, const float* __restrict__ xmask,
    int* __restrict__ E_idx, float* __restrict__ D_nb)
{
  const int i = blockIdx.x;
  const int t = threadIdx.x;
  __shared__ float sd[128];
  __shared__ int   sj[128];
  __shared__ int   sel[KN];
  const float cx = X[(size_t)i*12 + 3];
  const float cy = X[(size_t)i*12 + 4];
  const float cz = X[(size_t)i*12 + 5];
  const float mi = xmask[i];
  for (int r = 0; r < KN; ++r) {
    float best = 3.0e38f; int bestj = 0x7fffffff;
    for (int j = t; j < Nn; j += 128) {
      bool skip = false;
      for (int q = 0; q < r; ++q) if (sel[q] == j) { skip = true; break; }
      if (skip) continue;
      const float dx = X[(size_t)j*12+3] - cx;
      const float dy = X[(size_t)j*12+4] - cy;
      const float dz = X[(size_t)j*12+5] - cz;
      float d = sqrtf(fmaxf(dx*dx + dy*dy + dz*dz, EPSf));
      d += (1.0f - mi * xmask[j]) * 1.0e6f;
      if (d < best || (d == best && j < bestj)) { best = d; bestj = j; }
    }
    sd[t] = best; sj[t] = bestj;
    __syncthreads();
    for (int s = 64; s > 0; s >>= 1) {
      if (t < s) {
        if (sd[t+s] < sd[t] || (sd[t+s] == sd[t] && sj[t+s] < sj[t])) {
          sd[t] = sd[t+s]; sj[t] = sj[t+s];
        }
      }
      __syncthreads();
    }
    if (t == 0) {
      sel[r] = sj[0];
      E_idx[(size_t)i*KN + r] = sj[0];
      D_nb [(size_t)i*KN + r] = sd[0];
    }
    __syncthreads();
  }
}

__global__ __launch_bounds__(256) void maskE_kernel(
    const int* __restrict__ E_idx, const float* __restrict__ xmask,
    float* __restrict__ maskE)
{
  const int e = blockIdx.x * blockDim.x + threadIdx.x;
  if (e >= Esz) return;
  maskE[e] = xmask[e / KN] * xmask[E_idx[e]];
}

// ---------------------------------------------------------------------------
// Node features: Vv = [f, b, side, od] (channel-major, ld=4), dihedral
// scalars -> LN -> nodeS[:,0:50];  V_term copied -> nodeS[:,50:114].
// ---------------------------------------------------------------------------
__global__ __launch_bounds__(256) void node_feat_kernel(
    const float* __restrict__ X, const float* __restrict__ Vterm,
    const float* __restrict__ Wn, const float* __restrict__ bn,
    const float* __restrict__ lnVg, const float* __restrict__ lnVb,
    float* __restrict__ Vv, float* __restrict__ nodeS)
{
  const int i = blockIdx.x * blockDim.x + threadIdx.x;
  if (i >= Nn) return;
  float nA[3], caA[3], cA[3], oA[3];
  for (int c = 0; c < 3; ++c) {
    nA[c]  = X[(size_t)i*12 + 0 + c];
    caA[c] = X[(size_t)i*12 + 3 + c];
    cA[c]  = X[(size_t)i*12 + 6 + c];
    oA[c]  = X[(size_t)i*12 + 9 + c];
  }
  float f[3] = {0,0,0}, bv[3] = {0,0,0};
  if (i + 1 < Nn) { for (int c=0;c<3;++c) f[c]  = X[(size_t)(i+1)*12+3+c] - caA[c]; norm3(f); }
  if (i >= 1)     { for (int c=0;c<3;++c) bv[c] = X[(size_t)(i-1)*12+3+c] - caA[c]; norm3(bv); }
  float nd[3], cd[3], bis[3], perp[3], side[3], od[3];
  for (int c = 0; c < 3; ++c) { nd[c]=nA[c]-caA[c]; cd[c]=cA[c]-caA[c]; od[c]=oA[c]-caA[c]; }
  norm3(nd); norm3(cd); norm3(od);
  for (int c = 0; c < 3; ++c) bis[c] = nd[c] + cd[c];
  norm3(bis);
  cross3(nd, cd, perp); norm3(perp);
  const float s13 = 0.5773502691896258f;  // sqrt(1/3)
  const float s23 = 0.8164965809277260f;  // sqrt(2/3)
  for (int c = 0; c < 3; ++c) side[c] = -bis[c]*s13 - perp[c]*s23;
  for (int c = 0; c < 3; ++c) {
    float* row = Vv + ((size_t)i*3 + c) * 4;
    row[0] = f[c]; row[1] = bv[c]; row[2] = side[c]; row[3] = od[c];
  }
  // backbone dihedrals
  float dih[3];
  const int TN = 3 * Nn;
  for (int d = 0; d < 3; ++d) {
    const int tpos = 3*i + d;
    float val = 0.f;
    if (tpos >= 1 && tpos < TN - 2) {
      const int m = tpos - 1;
      float u2[3], u1[3], u0[3], n2[3], n1v[3];
      udir(X, m, u2); udir(X, m+1, u1); udir(X, m+2, u0);
      cross3(u2, u1, n2);  norm3(n2);
      cross3(u1, u0, n1v); norm3(n1v);
      float cc = n2[0]*n1v[0] + n2[1]*n1v[1] + n2[2]*n1v[2];
      cc = fminf(fmaxf(cc, -1.f + 1e-7f), 1.f - 1e-7f);
      const float sgdot = u2[0]*n1v[0] + u2[1]*n1v[1] + u2[2]*n1v[2];
      const float sg = (sgdot > 0.f) ? 1.f : ((sgdot < 0.f) ? -1.f : 0.f);
      val = sg * acosf(cc);
    }
    dih[d] = val;
  }
  const float feat6[6] = { cosf(dih[0]), cosf(dih[1]), cosf(dih[2]),
                           sinf(dih[0]), sinf(dih[1]), sinf(dih[2]) };
  float s[HS];
  for (int o = 0; o < HS; ++o) {
    float a = bn[o];
    for (int q = 0; q < 6; ++q) a += feat6[q] * Wn[q*HS + o];
    s[o] = a;
  }
  float mu = 0.f;  for (int o = 0; o < HS; ++o) mu += s[o];  mu /= HS;
  float var = 0.f; for (int o = 0; o < HS; ++o) { float d0 = s[o]-mu; var += d0*d0; } var /= HS;
  const float inv = rsqrtf(var + EPSf);
  float* outS = nodeS + (size_t)i * (HS + IN_DIM);
  for (int o = 0; o < HS; ++o) outS[o] = (s[o]-mu)*inv*lnVg[o] + lnVb[o];
  for (int t2 = 0; t2 < IN_DIM; ++t2) outS[HS + t2] = Vterm[(size_t)i*IN_DIM + t2];
}

// ---------------------------------------------------------------------------
// Edge features: Ev unit vector (ld=1), [rbf16|pe16]@We + LN -> edgeS[:,0:32],
// gathered E_term -> edgeS[:,32:96].
// ---------------------------------------------------------------------------
__global__ __launch_bounds__(256) void edge_feat_kernel(
    const float* __restrict__ X, const float* __restrict__ Eterm,
    const int* __restrict__ chain, const int* __restrict__ E_idx,
    const float* __restrict__ D_nb,
    const float* __restrict__ We, const float* __restrict__ be,
    const float* __restrict__ lnEg, const float* __restrict__ lnEb,
    float* __restrict__ Ev, float* __restrict__ edgeS)
{
  const int e = blockIdx.x * blockDim.x + threadIdx.x;
  if (e >= Esz) return;
  const int i = e / KN;
  const int j = E_idx[e];
  float dv[3];
  for (int c = 0; c < 3; ++c) dv[c] = X[(size_t)j*12+3+c] - X[(size_t)i*12+3+c];
  norm3(dv);
  for (int c = 0; c < 3; ++c) Ev[(size_t)e*3 + c] = dv[c];
  float in32[32];
  const float d = D_nb[e];
  for (int q = 0; q < 16; ++q) {
    const float muq = 2.0f + (20.0f/15.0f) * q;   // linspace(2,22,16)
    const float z = (d - muq) / 1.25f;            // sigma = 20/16
    in32[q] = expf(-z*z);
  }
  float off = (float)(j - i);
  if (chain[j] != chain[i]) off = 0.f;
  for (int q = 0; q < 8; ++q) {
    const float fr = expf(-9.210340371976184f * (float)q / 8.0f); // 1e4^(-q/8)
    const float ang = off * fr;
    in32[16 + q] = cosf(ang);
    in32[24 + q] = sinf(ang);
  }
  float s[32];
  for (int o = 0; o < 32; ++o) {
    float a = be[o];
    for (int q = 0; q < 32; ++q) a += in32[q] * We[q*32 + o];
    s[o] = a;
  }
  float mu = 0.f;  for (int o = 0; o < 32; ++o) mu += s[o];  mu /= 32.f;
  float var = 0.f; for (int o = 0; o < 32; ++o) { float d0 = s[o]-mu; var += d0*d0; } var /= 32.f;
  const float inv = rsqrtf(var + EPSf);
  float* outS = edgeS + (size_t)e * 96;
  for (int o = 0; o < 32; ++o) outS[o] = (s[o]-mu)*inv*lnEg[o] + lnEb[o];
  const float* et = Eterm + ((size_t)i*Nn + j) * IN_DIM;
  for (int t2 = 0; t2 < IN_DIM; ++t2) outS[32 + t2] = et[t2];
}

// ---------------------------------------------------------------------------
// s_cat[m] = [ S_in[m, 0:si] , ||VH[m,:,j]||_clip  j=0..h-1 ]
// ---------------------------------------------------------------------------
__global__ __launch_bounds__(256) void concat_norm_kernel(
    float* __restrict__ SCAT, const float* __restrict__ Sin, int ldSin,
    const float* __restrict__ VH, int h, int si, int M)
{
  const int W = si + h;
  const int idx = blockIdx.x * blockDim.x + threadIdx.x;
  if (idx >= M * W) return;
  const int m = idx / W, c = idx % W;
  float v;
  if (c < si) {
    v = Sin[(size_t)m * ldSin + c];
  } else {
    const int j = c - si;
    const float a0 = VH[((size_t)m*3 + 0)*h + j];
    const float a1 = VH[((size_t)m*3 + 1)*h + j];
    const float a2 = VH[((size_t)m*3 + 2)*h + j];
    v = sqrtf(fmaxf(a0*a0 + a1*a1 + a2*a2, EPSf));
  }
  SCAT[(size_t)m * W + c] = v;
}

// v *= sigmoid(||v||_clip)  (per vector)
__global__ __launch_bounds__(256) void vgate_kernel(float* __restrict__ V, int vo, int M)
{
  const int idx = blockIdx.x * blockDim.x + threadIdx.x;
  if (idx >= M * vo) return;
  const int m = idx / vo, v = idx % vo;
  float a0 = V[((size_t)m*3 + 0)*vo + v];
  float a1 = V[((size_t)m*3 + 1)*vo + v];
  float a2 = V[((size_t)m*3 + 2)*vo + v];
  const float n = sqrtf(fmaxf(a0*a0 + a1*a1 + a2*a2, EPSf));
  const float g = 1.f / (1.f + expf(-n));
  V[((size_t)m*3 + 0)*vo + v] = a0 * g;
  V[((size_t)m*3 + 1)*vo + v] = a1 * g;
  V[((size_t)m*3 + 2)*vo + v] = a2 * g;
}

// mv = [hEv | hVv_i | hVv_j] (24 vecs, channel-major), ms = [hEs|hVs_i|hVs_j]
__global__ __launch_bounds__(256) void cat_endpoints_kernel(
    const float* __restrict__ hEv, const float* __restrict__ hEs,
    const float* __restrict__ hVv, const float* __restrict__ hVs,
    const int* __restrict__ E_idx,
    float* __restrict__ mv, float* __restrict__ ms)
{
  const int e = blockIdx.x * blockDim.x + threadIdx.x;
  if (e >= Esz) return;
  const int i = e / KN;
  const int j = E_idx[e];
  for (int c = 0; c < 3; ++c) {
    float* dst = mv + ((size_t)e*3 + c) * 24;
    const float* se  = hEv + ((size_t)e*3 + c) * HV;
    const float* si_ = hVv + ((size_t)i*3 + c) * HV;
    const float* sj_ = hVv + ((size_t)j*3 + c) * HV;
    for (int v = 0; v < HV; ++v) { dst[v] = se[v]; dst[8+v] = si_[v]; dst[16+v] = sj_[v]; }
  }
  float* d2 = ms + (size_t)e * 150;
  const float* a0 = hEs + (size_t)e * HS;
  const float* a1 = hVs + (size_t)i * HS;
  const float* a2 = hVs + (size_t)j * HS;
  for (int t2 = 0; t2 < HS; ++t2) { d2[t2] = a0[t2]; d2[50+t2] = a1[t2]; d2[100+t2] = a2[t2]; }
}

// node layer: masked mean over K, residual, GVP layernorm (in place on hV)
__global__ __launch_bounds__(128) void node_reduce_ln_kernel(
    float* __restrict__ hVv, float* __restrict__ hVs,
    const float* __restrict__ mvO, const float* __restrict__ msO,
    const int* __restrict__ E_idx, const float* __restrict__ xmask,
    const float* __restrict__ g, const float* __restrict__ bta)
{
  const int i = blockIdx.x * blockDim.x + threadIdx.x;
  if (i >= Nn) return;
  const float mi = xmask[i];
  float nv[3][HV];
  for (int c = 0; c < 3; ++c) for (int v = 0; v < HV; ++v) nv[c][v] = 0.f;
  float ns[HS];
  for (int t = 0; t < HS; ++t) ns[t] = 0.f;
  for (int k = 0; k < KN; ++k) {
    const int e = i * KN + k;
    const float mk = mi * xmask[E_idx[e]];
    for (int c = 0; c < 3; ++c) {
      const float* src = mvO + ((size_t)e*3 + c) * HV;
      for (int v = 0; v < HV; ++v) nv[c][v] += src[v] * mk;
    }
    const float* ss = msO + (size_t)e * HS;
    for (int t = 0; t < HS; ++t) ns[t] += ss[t] * mk;
  }
  const float invK = 1.0f / (float)KN;
  for (int c = 0; c < 3; ++c) for (int v = 0; v < HV; ++v)
    nv[c][v] = hVv[((size_t)i*3 + c)*HV + v] + nv[c][v] * invK;
  for (int t = 0; t < HS; ++t) ns[t] = hVs[(size_t)i*HS + t] + ns[t] * invK;
  float mu = 0.f;  for (int t = 0; t < HS; ++t) mu += ns[t];  mu /= HS;
  float var = 0.f; for (int t = 0; t < HS; ++t) { float d0 = ns[t]-mu; var += d0*d0; } var /= HS;
  const float inv = rsqrtf(var + EPSf);
  for (int t = 0; t < HS; ++t) hVs[(size_t)i*HS + t] = (ns[t]-mu)*inv*g[t] + bta[t];
  float vn = 0.f;
  for (int c = 0; c < 3; ++c) for (int v = 0; v < HV; ++v) vn += nv[c][v]*nv[c][v];
  vn /= (float)HV;
  const float den = rsqrtf(fmaxf(vn, EPSf));
  for (int c = 0; c < 3; ++c) for (int v = 0; v < HV; ++v)
    hVv[((size_t)i*3 + c)*HV + v] = nv[c][v] * den;
}

// generic residual + GVP layernorm (+ optional row mask), in place
__global__ __launch_bounds__(128) void residual_ln_kernel(
    float* __restrict__ Vst, float* __restrict__ Sst,
    const float* __restrict__ dV, const float* __restrict__ dS,
    const float* __restrict__ g, const float* __restrict__ bta,
    const float* __restrict__ rowmask, int M)
{
  const int m = blockIdx.x * blockDim.x + threadIdx.x;
  if (m >= M) return;
  float nv[3][HV];
  for (int c = 0; c < 3; ++c) for (int v = 0; v < HV; ++v)
    nv[c][v] = Vst[((size_t)m*3 + c)*HV + v] + dV[((size_t)m*3 + c)*HV + v];
  float ns[HS];
  for (int t = 0; t < HS; ++t) ns[t] = Sst[(size_t)m*HS + t] + dS[(size_t)m*HS + t];
  float mu = 0.f;  for (int t = 0; t < HS; ++t) mu += ns[t];  mu /= HS;
  float var = 0.f; for (int t = 0; t < HS; ++t) { float d0 = ns[t]-mu; var += d0*d0; } var /= HS;
  const float inv = rsqrtf(var + EPSf);
  const float mk = rowmask ? rowmask[m] : 1.f;
  for (int t = 0; t < HS; ++t)
    Sst[(size_t)m*HS + t] = ((ns[t]-mu)*inv*g[t] + bta[t]) * mk;
  float vn = 0.f;
  for (int c = 0; c < 3; ++c) for (int v = 0; v < HV; ++v) vn += nv[c][v]*nv[c][v];
  vn /= (float)HV;
  const float den = rsqrtf(fmaxf(vn, EPSf));
  for (int c = 0; c < 3; ++c) for (int v = 0; v < HV; ++v)
    Vst[((size_t)m*3 + c)*HV + v] = nv[c][v] * den * mk;
}

// reverse-edge position (first match, -1 if none)
__global__ __launch_bounds__(256) void pair_pos_kernel(
    const int* __restrict__ E_idx, int* __restrict__ pairPos)
{
  const int e = blockIdx.x * blockDim.x + threadIdx.x;
  if (e >= Esz) return;
  const int i = e / KN, j = E_idx[e];
  int pos = -1;
  for (int r = 0; r < KN; ++r)
    if (E_idx[(size_t)j*KN + r] == i) { pos = r; break; }
  pairPos[e] = pos;
}

// symmetric 20x20 merge of duplicate pair energies
__global__ __launch_bounds__(256) void merge_kernel(
    const float* __restrict__ he, const int* __restrict__ E_idx,
    const int* __restrict__ pairPos, float* __restrict__ out)
{
  const size_t idx = (size_t)blockIdx.x * blockDim.x + threadIdx.x;
  if (idx >= (size_t)Esz * OUT_DIM) return;
  const int e = (int)(idx / OUT_DIM);
  const int t = (int)(idx % OUT_DIM);
  const int a = t / 20, b2 = t % 20;
  float v = he[(size_t)e*OUT_DIM + t];
  const int p = pairPos[e];
  if (p >= 0) {
    const int j = E_idx[e];
    const float rv = he[((size_t)j*KN + p)*OUT_DIM + b2*20 + a];
    v = 0.5f * (v + rv);
  }
  out[idx] = v;
}

__global__ __launch_bounds__(256) void eidx_out_kernel(
    const int* __restrict__ E_idx, int* __restrict__ out)
{
  const int e = blockIdx.x * blockDim.x + threadIdx.x;
  if (e >= Esz) return;
  out[e] = E_idx[e];
}

// ---------------------------------------------------------------------------
// host-side GVP driver
// ---------------------------------------------------------------------------
struct GVPW {
  const float *bs, *wh, *ws, *wv;
  int vi, vo, si, so, h;
};

static inline GVPW gvp_at(void* const* din, int base, int vi, int vo, int si, int so) {
  GVPW w;
  w.bs = (const float*)din[base + 0];
  w.wh = (const float*)din[base + 1];
  w.ws = (const float*)din[base + 2];
  w.wv = (vo > 0) ? (const float*)din[base + 3] : nullptr;
  w.vi = vi; w.vo = vo; w.si = si; w.so = so; w.h = (vi > vo) ? vi : vo;
  return w;
}

static inline void launch_gemm(hipStream_t st, const float* A, int lda,
                               const float* W, int ldw, float* C, int ldc,
                               const float* bias, int M, int K, int Ncols, int act)
{
  const int tilesM  = (M + 15) / 16;
  const int stripsM = (tilesM + 3) / 4;
  const int nWork   = stripsM * ((Ncols + 15) / 16);
  const int blocks  = (nWork + 7) / 8;
  gemm_f32_wmma<<<blocks, 256, 0, st>>>(A, lda, W, ldw, C, ldc, bias, M, K, Ncols, act);
}

// One full GVP: VH = Vin@wh ; SCAT = [Sin || ||VH||] ; Sout = SCAT@ws+bs (relu?) ;
// Vout = VH@wv (sigmoid gate?).  Vout may alias Vin, Sout may alias Sin.
static void run_gvp(hipStream_t st, int M,
                    const float* Vin, int ldVin, const float* Sin, int ldSin,
                    float* VH, float* SCAT, float* Vout, float* Sout,
                    const GVPW& w, bool act)
{
  launch_gemm(st, Vin, ldVin, w.wh, w.h, VH, w.h, nullptr, 3*M, w.vi, w.h, 0);
  {
    const int tot = M * (w.si + w.h);
    concat_norm_kernel<<<(tot + 255)/256, 256, 0, st>>>(SCAT, Sin, ldSin, VH, w.h, w.si, M);
  }
  launch_gemm(st, SCAT, w.si + w.h, w.ws, w.so, Sout, w.so, w.bs,
              M, w.si + w.h, w.so, act ? 1 : 0);
  if (w.vo > 0) {
    launch_gemm(st, VH, w.h, w.wv, w.vo, Vout, w.vo, nullptr, 3*M, w.h, w.vo, 0);
    if (act) {
      const int tot = M * w.vo;
      vgate_kernel<<<(tot + 255)/256, 256, 0, st>>>(Vout, w.vo, M);
    }
  }
}

} // anonymous namespace

// ---------------------------------------------------------------------------
// kernel_launch
// d_in order (jax tree-flatten of setup_inputs, dict keys sorted):
//   0:V_term 1:E_term 2:X 3:x_mask 4:chain_idx, then params leaves:
//   W_e{bs,wh,ws,wv} W_out{bs,wh,ws} W_v{bs,wh,ws,wv}
//   feat{We,Wn,be,bn,lnE.b,lnE.g,lnV.b,lnV.g}
//   layers[0..2]{edge,node}{ff0{4},ff1{4},msg0{4},msg1{4},msg2{4},n0.b,n0.g,n1.b,n1.g}
// d_out = hE (Esz*400 f32) then E_idx (Esz i32).
// ---------------------------------------------------------------------------
extern "C" void kernel_launch(void* const* d_in, const int* in_sizes, int n_in,
                              void* d_out, int out_size, void* d_ws, size_t ws_size,
                              hipStream_t stream)
{
  (void)in_sizes; (void)n_in; (void)out_size; (void)ws_size;

  const float* V_term = (const float*)d_in[0];
  const float* E_term = (const float*)d_in[1];
  const float* X      = (const float*)d_in[2];
  const float* xmask  = (const float*)d_in[3];
  const int*   chain  = (const int*)d_in[4];

  const int P = 5;
  const GVPW W_e   = gvp_at(d_in, P + 0, 1, 8, 96, 50);
  const GVPW W_out = gvp_at(d_in, P + 4, 8, 0, 50, 400);   // 3 leaves (no wv)
  const GVPW W_v   = gvp_at(d_in, P + 7, 4, 8, 114, 50);
  const float* fWe  = (const float*)d_in[P + 11];
  const float* fWn  = (const float*)d_in[P + 12];
  const float* fbe  = (const float*)d_in[P + 13];
  const float* fbn  = (const float*)d_in[P + 14];
  const float* lnEb = (const float*)d_in[P + 15];
  const float* lnEg = (const float*)d_in[P + 16];
  const float* lnVb = (const float*)d_in[P + 17];
  const float* lnVg = (const float*)d_in[P + 18];
  const int LBASE = P + 19;                                // 24

  // ---- workspace carve-up (all 4-byte aligned)
  int* E_idx   = (int*)d_ws;
  int* pairPos = E_idx + Esz;
  float* fb = (float*)(pairPos + Esz);
  float* D_nb   = fb; fb += Esz;
  float* maskE  = fb; fb += Esz;
  float* hVv    = fb; fb += (size_t)Nn * 3 * HV;
  float* hVs    = fb; fb += (size_t)Nn * HS;
  float* hEv    = fb; fb += (size_t)Esz * 3 * HV;
  float* hEs    = fb; fb += (size_t)Esz * HS;
  float* nodeV1 = fb; fb += (size_t)Nn * 3 * 32;
  float* nodeV2 = fb; fb += (size_t)Nn * 3 * 32;
  float* nodeS1 = fb; fb += (size_t)Nn * 224;
  float* nodeS2 = fb; fb += (size_t)Nn * 224;
  float* bufV1  = fb; fb += (size_t)Esz * 3 * 32;
  float* bufV2  = fb; fb += (size_t)Esz * 3 * 32;
  float* bufS1  = fb; fb += (size_t)Esz * 448;   // wide: also holds [E][400] W_out result
  float* bufS2  = fb; fb += (size_t)Esz * 224;

  // ---- features
  topk_kernel<<<Nn, 128, 0, stream>>>(X, xmask, E_idx, D_nb);
  maskE_kernel<<<(Esz + 255)/256, 256, 0, stream>>>(E_idx, xmask, maskE);
  node_feat_kernel<<<(Nn + 255)/256, 256, 0, stream>>>(
      X, V_term, fWn, fbn, lnVg, lnVb, nodeV1, nodeS1);
  edge_feat_kernel<<<(Esz + 255)/256, 256, 0, stream>>>(
      X, E_term, chain, E_idx, D_nb, fWe, fbe, lnEg, lnEb, bufV1, bufS1);

  // ---- initial embeddings
  run_gvp(stream, Nn,  nodeV1, 4, nodeS1, 114, nodeV2, nodeS2, hVv, hVs, W_v, false);
  run_gvp(stream, Esz, bufV1,  1, bufS1,  96,  bufV2,  bufS2,  hEv, hEs, W_e, false);

  // ---- message-passing layers
  for (int L = 0; L < 3; ++L) {
    const int eb = LBASE + 48 * L;       // 'edge' block (alphabetical first)
    const int nb = eb + 24;              // 'node' block
    const GVPW nff0  = gvp_at(d_in, nb + 0,  8, 16, 50, 200);
    const GVPW nff1  = gvp_at(d_in, nb + 4,  16, 8, 200, 50);
    const GVPW nmsg0 = gvp_at(d_in, nb + 8,  24, 8, 150, 50);
    const GVPW nmsg1 = gvp_at(d_in, nb + 12, 8,  8, 50,  50);
    const GVPW nmsg2 = gvp_at(d_in, nb + 16, 8,  8, 50,  50);
    const float* n_n0b = (const float*)d_in[nb + 20];
    const float* n_n0g = (const float*)d_in[nb + 21];
    const float* n_n1b = (const float*)d_in[nb + 22];
    const float* n_n1g = (const float*)d_in[nb + 23];
    const GVPW eff0  = gvp_at(d_in, eb + 0,  8, 16, 50, 200);
    const GVPW eff1  = gvp_at(d_in, eb + 4,  16, 8, 200, 50);
    const GVPW emsg0 = gvp_at(d_in, eb + 8,  24, 8, 150, 50);
    const GVPW emsg1 = gvp_at(d_in, eb + 12, 8,  8, 50,  50);
    const GVPW emsg2 = gvp_at(d_in, eb + 16, 8,  8, 50,  50);
    const float* e_n0b = (const float*)d_in[eb + 20];
    const float* e_n0g = (const float*)d_in[eb + 21];
    const float* e_n1b = (const float*)d_in[eb + 22];
    const float* e_n1g = (const float*)d_in[eb + 23];

    // ---- node sub-layer
    cat_endpoints_kernel<<<(Esz + 255)/256, 256, 0, stream>>>(
        hEv, hEs, hVv, hVs, E_idx, bufV1, bufS1);
    run_gvp(stream, Esz, bufV1, 24, bufS1, 150, bufV2, bufS2, bufV1, bufS1, nmsg0, true);
    run_gvp(stream, Esz, bufV1, 8,  bufS1, 50,  bufV2, bufS2, bufV1, bufS1, nmsg1, true);
    run_gvp(stream, Esz, bufV1, 8,  bufS1, 50,  bufV2, bufS2, bufV1, bufS1, nmsg2, false);
    node_reduce_ln_kernel<<<(Nn + 127)/128, 128, 0, stream>>>(
        hVv, hVs, bufV1, bufS1, E_idx, xmask, n_n0g, n_n0b);
    run_gvp(stream, Nn, hVv,    8,  hVs,    50,  nodeV2, nodeS2, nodeV1, nodeS1, nff0, true);
    run_gvp(stream, Nn, nodeV1, 16, nodeS1, 200, nodeV2, nodeS2, nodeV1, nodeS1, nff1, false);
    residual_ln_kernel<<<(Nn + 127)/128, 128, 0, stream>>>(
        hVv, hVs, nodeV1, nodeS1, n_n1g, n_n1b, xmask, Nn);

    // ---- edge sub-layer
    cat_endpoints_kernel<<<(Esz + 255)/256, 256, 0, stream>>>(
        hEv, hEs, hVv, hVs, E_idx, bufV1, bufS1);
    run_gvp(stream, Esz, bufV1, 24, bufS1, 150, bufV2, bufS2, bufV1, bufS1, emsg0, true);
    run_gvp(stream, Esz, bufV1, 8,  bufS1, 50,  bufV2, bufS2, bufV1, bufS1, emsg1, true);
    run_gvp(stream, Esz, bufV1, 8,  bufS1, 50,  bufV2, bufS2, bufV1, bufS1, emsg2, false);
    residual_ln_kernel<<<(Esz + 127)/128, 128, 0, stream>>>(
        hEv, hEs, bufV1, bufS1, e_n0g, e_n0b, nullptr, Esz);
    run_gvp(stream, Esz, hEv,   8,  hEs,   50,  bufV2, bufS2, bufV1, bufS1, eff0, true);
    run_gvp(stream, Esz, bufV1, 16, bufS1, 200, bufV2, bufS2, bufV1, bufS1, eff1, false);
    residual_ln_kernel<<<(Esz + 127)/128, 128, 0, stream>>>(
        hEv, hEs, bufV1, bufS1, e_n1g, e_n1b, maskE, Esz);
  }

  // ---- output projection (vo=0: scalar GEMM only) -> bufS1 [E][400]
  run_gvp(stream, Esz, hEv, 8, hEs, 50, bufV2, bufS2, bufV1, bufS1, W_out, false);

  // ---- symmetric pair merge + outputs
  pair_pos_kernel<<<(Esz + 255)/256, 256, 0, stream>>>(E_idx, pairPos);
  {
    const size_t tot = (size_t)Esz * OUT_DIM;
    merge_kernel<<<(unsigned)((tot + 255)/256), 256, 0, stream>>>(
        bufS1, E_idx, pairPos, (float*)d_out);
  }
  eidx_out_kernel<<<(Esz + 255)/256, 256, 0, stream>>>(
      E_idx, (int*)d_out + (size_t)Esz * OUT_DIM);
}